// NEMON_584115552274
// MI455X (gfx1250) — compile-verified
//
#include <hip/hip_runtime.h>

// NEMON fused monotone-residual layer for MI455X (gfx1250).
//
//   out = x @ U_w^T + U_b + ((z*dw) @ A_w^T)/dw + z*(0.05 - rowsum|A_w|)
//       = [x | z] @ Wcat^T + U_b            (K = 256 fused GEMM)
// where Wcat[o, 0:128]   = U_w[o,:]
//       Wcat[o, 128+j]   = A_w[o,j]*dw[j]/dw[o] + (o==j)*(0.05 - rowsum_o)
//
// HBM floor: 402 MB moved @ 23.3 TB/s ~= 17.3 us; fp32 V_WMMA_F32_16X16X4_F32
// keeps reference precision while staying on the matrix pipe. Each wave
// register-blocks 32x64 (2 M-tiles x 4 N-tiles) so every LDS fragment feeds
// 2-4 WMMAs: 6 ds_load_2addr_b64 per 8 WMMAs instead of 9.

typedef float v2f __attribute__((ext_vector_type(2)));
typedef float v8f __attribute__((ext_vector_type(8)));

#define IN_F     128
#define OUT_F    128
#define K_TOT    256
#define ROWS_BLK 128
#define XS       260          // padded LDS row stride (floats); 260 % 64 == 4 -> conflict-free frag loads
#define M_MARGIN 0.05f

// ---------------------------------------------------------------------------
// Prep: fold U_w, transformed A_w, diagonal terms into one [128 x 256] weight.
// One block per output row o; 128 threads over j.
// ---------------------------------------------------------------------------
__global__ __launch_bounds__(128, 4) void nemon_prep(const float* __restrict__ U_w,
                                                     const float* __restrict__ A_w,
                                                     const float* __restrict__ d,
                                                     float* __restrict__ Wt) {
  const int o = blockIdx.x;
  const int j = threadIdx.x;
  __shared__ float red[128];

  const float a = A_w[o * OUT_F + j];
  red[j] = fabsf(a);
  __syncthreads();
  #pragma unroll
  for (int s = 64; s > 0; s >>= 1) {
    if (j < s) red[j] += red[j + s];
    __syncthreads();
  }
  const float row_sum = red[0];

  const float w_scale = __expf(d[j]) * __expf(-d[o]);   // dw[j] / dw[o]
  Wt[o * K_TOT + j] = U_w[o * IN_F + j];
  float w2 = a * w_scale;
  if (j == o) w2 += M_MARGIN - row_sum;
  Wt[o * K_TOT + IN_F + j] = w2;
}

// ---------------------------------------------------------------------------
// Main fused GEMM: C[128rows x 128cols] per block, K=256, fp32 WMMA.
// 256 threads = 8 wave32s arranged 4(row-groups) x 2(col-groups);
// each wave owns a 32(M) x 64(N) register tile = 8 accumulator tiles.
// LDS: sX = [x|z] tile (128 x 256, padded), sW = full weight (128 x 256, padded)
//      => 2 * 128 * 260 * 4 B = 260 KB of the WGP's 320 KB LDS.
// ---------------------------------------------------------------------------
__global__ __launch_bounds__(256, 1) void nemon_gemm(const float* __restrict__ x,
                                                     const float* __restrict__ z,
                                                     const float* __restrict__ Wt,
                                                     const float* __restrict__ U_b,
                                                     float* __restrict__ out) {
  extern __shared__ __align__(16) float smem[];
  float* sX = smem;                      // [ROWS_BLK][XS]  rows = batch rows, cols = K
  float* sW = smem + ROWS_BLK * XS;      // [OUT_F][XS]     rows = output col n, cols = K

  const int tid  = threadIdx.x;
  const int lane = tid & 31;
  const int wave = tid >> 5;
  const size_t rowBase = (size_t)blockIdx.x * ROWS_BLK;

  // Pull the L2-resident fused weight toward this WGP (global_prefetch_b8).
  __builtin_prefetch(Wt + (tid << 7), 0, 0);   // 256 thr * 512 B = full 128 KB

  // ---- stage [x | z] tile, float4-coalesced ----
  const float4* x4 = (const float4*)(x + rowBase * IN_F);
  const float4* z4 = (const float4*)(z + rowBase * OUT_F);
  #pragma unroll 4
  for (int idx = tid; idx < ROWS_BLK * 32; idx += 256) {
    const int r = idx >> 5, c = idx & 31;
    *(float4*)&sX[r * XS + c * 4]        = x4[r * 32 + c];
    *(float4*)&sX[r * XS + IN_F + c * 4] = z4[r * 32 + c];
  }
  // ---- stage fused weight (L2-resident, 128 KB) ----
  const float4* w4 = (const float4*)Wt;
  #pragma unroll 4
  for (int idx = tid; idx < OUT_F * 64; idx += 256) {
    const int n = idx >> 6, c = idx & 63;
    *(float4*)&sW[n * XS + c * 4] = w4[n * 64 + c];
  }
  __syncthreads();

  // ---- per-wave 32(M) x 64(N), K=256 via V_WMMA_F32_16X16X4_F32 ----
  // A frag (16x4 f32): lanes 0-15 carry K = k0,k0+1 for row M=lane,
  //                    lanes 16-31 carry K = k0+2,k0+3 for row M=lane-16.
  const int rowGroup = wave & 3;            // 4 groups of 32 rows
  const int colGroup = wave >> 2;           // 2 groups of 64 cols
  const int khalf = 2 * (lane >> 4);
  const int ncol  = lane & 15;
  const int m0    = rowGroup * 32 + ncol;   // A-frag row, first M-tile
  const int nbase = colGroup * 64;

  v8f acc[2][4];
  const v8f vzero = {0.f, 0.f, 0.f, 0.f, 0.f, 0.f, 0.f, 0.f};
  #pragma unroll
  for (int h = 0; h < 2; ++h)
    #pragma unroll
    for (int t = 0; t < 4; ++t) acc[h][t] = vzero;

  for (int k0 = 0; k0 < K_TOT; k0 += 4) {
    const v2f a0 = *(const v2f*)&sX[m0 * XS + k0 + khalf];
    const v2f a1 = *(const v2f*)&sX[(m0 + 16) * XS + k0 + khalf];
    #pragma unroll
    for (int t = 0; t < 4; ++t) {
      const v2f b = *(const v2f*)&sW[(nbase + t * 16 + ncol) * XS + k0 + khalf];
      // 8 args: (neg_a, A, neg_b, B, c_mod, C, reuse_a, reuse_b)
      acc[0][t] = __builtin_amdgcn_wmma_f32_16x16x4_f32(
          false, a0, false, b, (short)0, acc[0][t], false, false);
      acc[1][t] = __builtin_amdgcn_wmma_f32_16x16x4_f32(
          false, a1, false, b, (short)0, acc[1][t], false, false);
    }
  }

  // ---- epilogue: + U_b, store. C/D layout: VGPR v -> M = v (lanes 0-15) / v+8
  // (lanes 16-31), lane%16 -> N within the 16-wide tile. ----
  const int rsel = 8 * (lane >> 4);
  #pragma unroll
  for (int h = 0; h < 2; ++h) {
    #pragma unroll
    for (int t = 0; t < 4; ++t) {
      const float bias = U_b[nbase + t * 16 + ncol];
      #pragma unroll
      for (int v = 0; v < 8; ++v) {
        const size_t r = rowBase + (size_t)(rowGroup * 32 + h * 16 + v + rsel);
        out[r * OUT_F + nbase + t * 16 + ncol] = acc[h][t][v] + bias;
      }
    }
  }
}

// ---------------------------------------------------------------------------
extern "C" void kernel_launch(void* const* d_in, const int* in_sizes, int n_in,
                              void* d_out, int out_size, void* d_ws, size_t ws_size,
                              hipStream_t stream) {
  const float* x   = (const float*)d_in[0];
  const float* z   = (const float*)d_in[1];
  const float* U_w = (const float*)d_in[2];
  const float* U_b = (const float*)d_in[3];
  const float* A_w = (const float*)d_in[4];
  const float* d   = (const float*)d_in[5];
  float* out = (float*)d_out;
  float* Wt  = (float*)d_ws;                      // 128 * 256 * 4 B = 128 KB scratch

  const int B = in_sizes[0] / IN_F;               // 262144
  const size_t ldsBytes = (size_t)2 * ROWS_BLK * XS * sizeof(float);  // 260 KB

  (void)hipFuncSetAttribute((const void*)nemon_gemm,
                            hipFuncAttributeMaxDynamicSharedMemorySize,
                            (int)ldsBytes);

  nemon_prep<<<OUT_F, 128, 0, stream>>>(U_w, A_w, d, Wt);
  nemon_gemm<<<B / ROWS_BLK, 256, ldsBytes, stream>>>(x, z, Wt, U_b, out);
}